// CenterQueryAttention_33225867002296
// MI455X (gfx1250) — compile-verified
//
#include <hip/hip_runtime.h>
#include <hip/hip_bf16.h>

// Problem constants (from reference setup_inputs)
constexpr int kB = 32;     // batch
constexpr int kC = 512;    // channels
constexpr int kL = 8192;   // sequence length
constexpr int kA = 128;    // attention dim

// Fused-pass tiling: 64 l-positions per tile, staged in LDS, double buffered.
constexpr int kTILE   = 64;            // l per tile
constexpr int kPITCH  = kTILE + 4;     // padded row pitch (floats) -> bank spread, 16B aligned
constexpr int kNT     = kL / kTILE;    // 128 tiles per batch
constexpr int kTPW    = 16;            // tiles processed per workgroup (pipelined)
constexpr int kChunks = kNT / kTPW;    // 8 chunks per batch -> grid 256 (~1 WG / WGP)
constexpr int kDynLDS = 2 * kC * kPITCH * 4; // 278528 B (CDNA5: up to 320KB/WG)

typedef __attribute__((ext_vector_type(2))) float v2f;
typedef __attribute__((ext_vector_type(8))) float v8f;

// ---------------------------------------------------------------------------
// Generic small-GEMM via V_WMMA_F32_16X16X4_F32 (unchanged from round 1;
// verified to lower to v_wmma_f32_16x16x4_f32). One 16x16 D tile per wave32.
// ---------------------------------------------------------------------------
__global__ __launch_bounds__(512) void wmma_gemm32(
    const float* __restrict__ Ab, const int* __restrict__ a_off_p,
    long long a_m, long long a_k,
    const float* __restrict__ Bb, long long b_k, long long b_n,
    float* __restrict__ Db, long long d_m, long long d_n,
    int tiles_m, int tiles_n, int K, float scale)
{
    const int wid  = (blockIdx.x * blockDim.x + threadIdx.x) >> 5;
    if (wid >= tiles_m * tiles_n) return;
    const int m_tile = wid / tiles_n;
    const int n_tile = wid % tiles_n;
    const int lane = threadIdx.x & 31;
    const int half = lane >> 4;
    const int l16  = lane & 15;

    const long long a_off = a_off_p ? (long long)a_off_p[0] : 0ll;
    const long long m = m_tile * 16 + l16;
    const long long n = n_tile * 16 + l16;
    const float* Arow = Ab + a_off + m * a_m;
    const float* Bcol = Bb + n * b_n;

    v8f acc = {};
    const int ksteps = K >> 2;
    #pragma unroll 4
    for (int kb = 0; kb < ksteps; ++kb) {
        const long long k0 = (long long)kb * 4 + 2 * half;
        v2f a, bfr;
        a.x   = Arow[k0 * a_k];
        a.y   = Arow[(k0 + 1) * a_k];
        bfr.x = Bcol[k0 * b_k];
        bfr.y = Bcol[(k0 + 1) * b_k];
        acc = __builtin_amdgcn_wmma_f32_16x16x4_f32(
            false, a, false, bfr, (short)0, acc, false, false);
    }
    #pragma unroll
    for (int r = 0; r < 8; ++r) {
        const long long row = m_tile * 16 + r + 8 * half;
        const long long col = n_tile * 16 + l16;
        Db[row * d_m + col * d_n] = acc[r] * scale;
    }
}

// ---------------------------------------------------------------------------
// Async global->LDS tile copy (CDNA5 ASYNCcnt path).
// Wave `wave` copies rows [wave*64, wave*64+64) of a 512-row x 64-float tile.
// One global_load_async_to_lds_b128 moves 16B per lane = 512B = 2 rows.
// ---------------------------------------------------------------------------
__device__ __forceinline__ void async_copy_tile(
    const float* __restrict__ x, int b, int l0, int wave, int lane, float* buf)
{
    const int half = lane >> 4;          // lanes 0-15: row 2i, 16-31: row 2i+1
    const int col  = (lane & 15) * 4;    // 4 floats = 16B per lane
    int row = wave * 64 + half;
    const float* g = x + ((size_t)b * kC + row) * (size_t)kL + l0 + col;
    unsigned lds = (unsigned)(size_t)(buf + row * kPITCH + col);
    #pragma unroll
    for (int i = 0; i < 32; ++i) {
        asm volatile("global_load_async_to_lds_b128 %0, %1, off"
                     :: "v"(lds), "v"(g) : "memory");
        g   += 2 * (size_t)kL;
        lds += 2 * kPITCH * 4;
    }
}

template <int N>
__device__ __forceinline__ void wait_asynccnt()
{
#if __has_builtin(__builtin_amdgcn_s_wait_asynccnt)
    __builtin_amdgcn_s_wait_asynccnt(N);
#else
    asm volatile("s_wait_asynccnt %0" :: "i"(N) : "memory");
#endif
}

// ---------------------------------------------------------------------------
// Fused single HBM pass over x, double-buffered through LDS:
//   prefetch tile t+1 (async DMA) while computing tile t from LDS.
//   phase A: attn[l]  = sum_c u[b,c] * x[b,c,l]
//   phase B: tile-local max / exp / sum-exp (flash-style)
//   phase C: ypart[c] = sum_l e[l] * x[b,c,l]    (from LDS, no re-read)
// ---------------------------------------------------------------------------
__global__ __launch_bounds__(256) void attn_pass_kernel(
    const float* __restrict__ x,    // [B,C,L]
    const float* __restrict__ u,    // [B,C] (1/sqrt(A) pre-folded)
    float* __restrict__ attn_raw,   // [B,L]
    float* __restrict__ mpart,      // [B*NT]
    float* __restrict__ spart,      // [B*NT]
    float* __restrict__ ypart)      // [B*NT, C]
{
    extern __shared__ float smem[];        // 2 tile buffers, kC x kPITCH each
    __shared__ float u_s[kC];
    __shared__ float part_s[256];
    __shared__ float e_s[kTILE];
    __shared__ float stat_s[1];

    float* buf0 = smem;
    float* buf1 = smem + kC * kPITCH;

    const int b     = blockIdx.x / kChunks;
    const int chunk = blockIdx.x % kChunks;
    const int t0    = chunk * kTPW;
    const int tid   = threadIdx.x;
    const int wave  = tid >> 5;
    const int lane  = tid & 31;

    for (int c = tid; c < kC; c += 256) u_s[c] = u[b * kC + c];

    async_copy_tile(x, b, t0 * kTILE, wave, lane, buf0);

    for (int i = 0; i < kTPW; ++i) {
        const int t = t0 + i;
        float* cur = (i & 1) ? buf1 : buf0;
        float* nxt = (i & 1) ? buf0 : buf1;
        if (i + 1 < kTPW) {
            async_copy_tile(x, b, (t + 1) * kTILE, wave, lane, nxt);
            wait_asynccnt<32>();   // in-order: tile t done, only t+1 in flight
        } else {
            wait_asynccnt<0>();
        }
        __syncthreads();

        // ---- phase A: 4 threads per l, each covers a quarter of C
        const int l = tid & 63;
        const int h = tid >> 6;
        float acc = 0.f;
        #pragma unroll 8
        for (int cc = 0; cc < kC / 4; ++cc) {
            const int c = h * (kC / 4) + cc;
            acc = fmaf(u_s[c], cur[c * kPITCH + l], acc);
        }
        part_s[tid] = acc;         // tid == h*64 + l
        __syncthreads();
        if (tid < kTILE) {
            float a = part_s[tid] + part_s[64 + tid] + part_s[128 + tid] + part_s[192 + tid];
            attn_raw[(size_t)b * kL + (size_t)t * kTILE + tid] = a;
            part_s[tid] = a;
        }
        __syncthreads();

        // ---- phase B: tile-local softmax stats (wave 0)
        if (wave == 0) {
            float m = fmaxf(part_s[lane], part_s[lane + 32]);
            #pragma unroll
            for (int off = 16; off > 0; off >>= 1) m = fmaxf(m, __shfl_xor(m, off, 32));
            if (lane == 0) stat_s[0] = m;
        }
        __syncthreads();
        const float mt = stat_s[0];
        if (tid < kTILE) e_s[tid] = __expf(part_s[tid] - mt);
        __syncthreads();
        if (wave == 0) {
            float s = e_s[lane] + e_s[lane + 32];
            #pragma unroll
            for (int off = 16; off > 0; off >>= 1) s += __shfl_xor(s, off, 32);
            if (lane == 0) { mpart[b * kNT + t] = mt; spart[b * kNT + t] = s; }
        }

        // ---- phase C: each wave owns 64 channel rows, all reads from LDS
        const int c0 = wave * 64;
        for (int r = 0; r < 64; ++r) {
            const int c = c0 + r;
            float a2 = fmaf(e_s[lane], cur[c * kPITCH + lane],
                            e_s[lane + 32] * cur[c * kPITCH + lane + 32]);
            #pragma unroll
            for (int off = 16; off > 0; off >>= 1) a2 += __shfl_xor(a2, off, 32);
            if (lane == 0) ypart[(size_t)(b * kNT + t) * kC + c] = a2;
        }
        __syncthreads();           // fences cur before it becomes next prefetch dest
    }
}

// ---------------------------------------------------------------------------
// Combine tile partials per batch: global max/sum-exp, rescale partial y.
// ---------------------------------------------------------------------------
__global__ __launch_bounds__(256) void combine_kernel(
    const float* __restrict__ mpart, const float* __restrict__ spart,
    const float* __restrict__ ypart,
    float* __restrict__ Mg, float* __restrict__ Sg, float* __restrict__ y)
{
    __shared__ float m_s[kNT], s_s[kNT], f_s[kNT];
    const int b = blockIdx.x;
    const int tid = threadIdx.x;
    for (int t = tid; t < kNT; t += 256) {
        m_s[t] = mpart[b * kNT + t];
        s_s[t] = spart[b * kNT + t];
    }
    __syncthreads();
    float M = m_s[0];
    for (int i = 1; i < kNT; ++i) M = fmaxf(M, m_s[i]);
    float S = 0.f;
    for (int i = 0; i < kNT; ++i) S += s_s[i] * __expf(m_s[i] - M);
    for (int t = tid; t < kNT; t += 256) f_s[t] = __expf(m_s[t] - M) / S;
    if (tid == 0) { Mg[b] = M; Sg[b] = S; }
    __syncthreads();
    for (int c = tid; c < kC; c += 256) {
        float acc = 0.f;
        for (int t = 0; t < kNT; ++t)
            acc = fmaf(ypart[(size_t)(b * kNT + t) * kC + c], f_s[t], acc);
        y[b * kC + c] = acc;
    }
}

// ---------------------------------------------------------------------------
// Emit normalized softmax weights w[b,l] (second output).
// ---------------------------------------------------------------------------
__global__ __launch_bounds__(256) void w_kernel(
    const float* __restrict__ attn_raw, const float* __restrict__ Mg,
    const float* __restrict__ Sg, float* __restrict__ w)
{
    const int i = blockIdx.x * blockDim.x + threadIdx.x;
    if (i >= kB * kL) return;
    const int b = i / kL;
    w[i] = __expf(attn_raw[i] - Mg[b]) / Sg[b];
}

extern "C" void kernel_launch(void* const* d_in, const int* in_sizes, int n_in,
                              void* d_out, int out_size, void* d_ws, size_t ws_size,
                              hipStream_t stream) {
    const float* x    = (const float*)d_in[0];
    const float* Wq   = (const float*)d_in[1];
    const float* Wk   = (const float*)d_in[2];
    const float* Wv   = (const float*)d_in[3];
    const float* Wout = (const float*)d_in[4];
    const int* center = (const int*)d_in[5];

    // Workspace layout (floats): ~9.7 MB total
    float* q     = (float*)d_ws;                  // [B,A]
    float* u     = q     + kB * kA;               // [B,C]
    float* attn  = u     + kB * kC;               // [B,L]
    float* mpart = attn  + (size_t)kB * kL;       // [B*NT]
    float* spart = mpart + kB * kNT;              // [B*NT]
    float* ypart = spart + kB * kNT;              // [B*NT,C]  (8 MB)
    float* Mg    = ypart + (size_t)kB * kNT * kC; // [B]
    float* Sg    = Mg    + kB;
    float* y     = Sg    + kB;                    // [B,C]
    float* tmat  = y     + kB * kC;               // [B,A]

    float* pooled = (float*)d_out;                // [B,C] first output
    float* w_out  = pooled + kB * kC;             // [B,L] second output

    const float inv_sqrt_a = 0.08838834764831845f; // 1/sqrt(128)

    // 1) q[b,a] = sum_c x[b,c,center] * Wq[a,c]   (M=32,N=128,K=512)
    wmma_gemm32<<<1, 512, 0, stream>>>(
        x, center, (long long)kC * kL, (long long)kL,
        Wq, 1ll, (long long)kC,
        q, (long long)kA, 1ll,
        2, 8, kC, 1.0f);

    // 2) u[b,c] = (1/sqrt(A)) * sum_a q[b,a] * Wk[a,c]   (M=32,N=512,K=128)
    wmma_gemm32<<<4, 512, 0, stream>>>(
        q, nullptr, (long long)kA, 1ll,
        Wk, (long long)kC, 1ll,
        u, (long long)kC, 1ll,
        2, 32, kA, inv_sqrt_a);

    // 3) Fused single HBM pass (async LDS double buffering)
    attn_pass_kernel<<<kB * kChunks, 256, kDynLDS, stream>>>(
        x, u, attn, mpart, spart, ypart);

    // 4) Merge tile partials per batch
    combine_kernel<<<kB, 256, 0, stream>>>(mpart, spart, ypart, Mg, Sg, y);

    // 5) Softmax weights output
    w_kernel<<<(kB * kL) / 256, 256, 0, stream>>>(attn, Mg, Sg, w_out);

    // 6) t[b,a] = sum_c y[b,c] * Wv[a,c]   (M=32,N=128,K=512)
    wmma_gemm32<<<1, 512, 0, stream>>>(
        y, nullptr, (long long)kC, 1ll,
        Wv, 1ll, (long long)kC,
        tmat, (long long)kA, 1ll,
        2, 8, kC, 1.0f);

    // 7) pooled[b,cc] = sum_a t[b,a] * Wout[cc,a]   (M=32,N=512,K=128)
    wmma_gemm32<<<4, 512, 0, stream>>>(
        tmat, nullptr, (long long)kA, 1ll,
        Wout, 1ll, (long long)kA,
        pooled, (long long)kC, 1ll,
        2, 32, kA, 1.0f);
}